// evalYolov2_18528488915299
// MI455X (gfx1250) — compile-verified
//
#include <hip/hip_runtime.h>
#include <cmath>

// YOLOv2 decode + stable sort + greedy NMS + filter, fused per image.
// Grid: one workgroup per batch image; 1024 threads (32 wave32 waves).
// gfx1250 paths: async global->LDS staging (ASYNCcnt), global_prefetch_b8,
// wave32 split barriers, LDS-resident bitonic sort + ds_add atomics.

#define OBJ_T 0.5f
#define NMS_T 0.45f
#define CLS_T 0.1f

#define NUMA  5
#define NCLS  80
#define ALEN  85          // 4 coords + 1 obj + 80 classes
#define HH    19
#define WW    19
#define HW    361         // 19*19
#define NDET  1805        // NUMA*HW
#define NPAD  2048        // next pow2 for bitonic sort
#define NTHREADS 1024
#define NPLANE (NUMA * 5)       // 25 coord/obj planes per image
#define NSTAGE (NPLANE * HW)    // 9025 staged floats (~36 KB)

#if defined(__gfx1250__) && __has_builtin(__builtin_amdgcn_global_load_async_to_lds_b32)
#define USE_ASYNC_LDS 1
typedef __attribute__((address_space(1))) int as1_int;
typedef __attribute__((address_space(3))) int as3_int;
#else
#define USE_ASYNC_LDS 0
#endif

__device__ __forceinline__ void wait_async_zero() {
#if USE_ASYNC_LDS
#if __has_builtin(__builtin_amdgcn_s_wait_asynccnt)
    __builtin_amdgcn_s_wait_asynccnt(0);
#else
    asm volatile("s_wait_asynccnt 0x0" ::: "memory");
#endif
#endif
}

__global__ __launch_bounds__(NTHREADS)
void yolov2_fused_kernel(const float* __restrict__ pred,
                         const float* __restrict__ anchors,
                         float* __restrict__ out)
{
    // Decoded detections (original cell order), padded with zeros.
    __shared__ float d_obj[NPAD], d_bx[NPAD], d_by[NPAD], d_bw[NPAD],
                     d_bh[NPAD], d_cid[NPAD], d_prob[NPAD];
    // Sort keys: (obj float bits << 32) | (NPAD-1 - idx) -> stable desc sort.
    __shared__ unsigned long long keys[NPAD];
    // Overlay: async staging buffer (Phase A only) vs sorted geometry (Phase C+).
    __shared__ union {
        float stage[NSTAGE];                                   // 36.1 KB
        struct {
            float x1[NPAD], x2[NPAD], y1[NPAD], y2[NPAD],
                  area[NPAD], obj[NPAD], s[NPAD];              // 57.3 KB
        } srt;
    } u;
    __shared__ int s_cnt;

    const int b   = blockIdx.x;
    const int tid = threadIdx.x;
    const float* img = pred + (size_t)b * (NUMA * ALEN * HW);

    // ---- Phase A1: async-stage the 25 coord/obj planes into LDS ----
    for (int e = tid; e < NSTAGE; e += NTHREADS) {
        int p  = e / HW;            // plane = a*5 + f
        int yx = e - p * HW;
        int a  = p / 5;
        int f  = p - a * 5;
        const float* g = img + (size_t)(a * ALEN + f) * HW + yx;
#if USE_ASYNC_LDS
        __builtin_amdgcn_global_load_async_to_lds_b32(
            (as1_int*)(int*)const_cast<float*>(g),
            (as3_int*)(int*)&u.stage[e],
            0, 0);
#else
        u.stage[e] = *g;
#endif
    }

    // ---- Phase A2: class max/argmax scan from global (overlaps async) ----
    for (int n = tid; n < NDET; n += NTHREADS) {
        int a  = n / HW;
        int yx = n - a * HW;
        const float* base = img + (size_t)(a * ALEN + 5) * HW + yx;
        __builtin_prefetch(base + 40 * HW, 0, 1);  // gfx1250 global_prefetch_b8
        float best = base[0];
        int   bc   = 0;
#pragma unroll 4
        for (int c = 1; c < NCLS; ++c) {
            float v = base[c * HW];
            if (v > best) { best = v; bc = c; }   // strict > keeps first max
        }
        d_prob[n] = best;            // raw class max (obj applied in A3)
        d_cid[n]  = (float)bc;
    }
    if (tid == 0) s_cnt = 0;
    wait_async_zero();               // own wave's async copies complete
    __syncthreads();                 // all waves' copies visible

    // ---- Phase A3: finalize decode from staged LDS ----
    for (int n = tid; n < NPAD; n += NTHREADS) {
        if (n < NDET) {
            int a  = n / HW;
            int yx = n - a * HW;
            int yy = yx / WW;
            int xx = yx - yy * WW;
            const float* cp = &u.stage[(a * 5) * HW + yx];
            float tx  = cp[0];
            float ty  = cp[HW];
            float tw  = cp[2 * HW];
            float th  = cp[3 * HW];
            float obj = cp[4 * HW];

            float bx = (tx + (float)xx) * (1.0f / WW);
            float by = (ty + (float)yy) * (1.0f / HH);
            float aw = anchors[2 * a];
            float ah = anchors[2 * a + 1];
            float bw = expf(tw) * (aw * (1.0f / WW));
            float bh = expf(th) * (ah * (1.0f / HH));

            d_obj[n] = obj; d_bx[n] = bx; d_by[n] = by;
            d_bw[n] = bw;   d_bh[n] = bh;
            d_prob[n] = d_prob[n] * obj;         // best * obj
            keys[n] = ((unsigned long long)__float_as_uint(obj) << 32)
                      | (unsigned)(NPAD - 1 - n);
        } else {
            d_obj[n] = 0.0f; d_bx[n] = 0.0f; d_by[n] = 0.0f;
            d_bw[n] = 0.0f;  d_bh[n] = 0.0f; d_cid[n] = 0.0f; d_prob[n] = 0.0f;
            keys[n] = (unsigned long long)(unsigned)(NPAD - 1 - n);
        }
    }
    __syncthreads();

    // ---- Phase B: bitonic sort (descending u64, stable via idx tiebreak) ----
    for (unsigned k = 2; k <= NPAD; k <<= 1) {
        for (unsigned j = k >> 1; j > 0; j >>= 1) {
            for (unsigned t = tid; t < NPAD; t += NTHREADS) {
                unsigned ixj = t ^ j;
                if (ixj > t) {
                    unsigned long long A  = keys[t];
                    unsigned long long Bv = keys[ixj];
                    bool desc = ((t & k) == 0);
                    bool sw = desc ? (A < Bv) : (A > Bv);
                    if (sw) { keys[t] = Bv; keys[ixj] = A; }
                }
            }
            __syncthreads();
        }
    }

    // ---- Phase C: gather sorted geometry (overwrites staging region) ----
    int localcnt = 0;
    for (int r = tid; r < NPAD; r += NTHREADS) {
        int n = (NPAD - 1) - (int)(unsigned)(keys[r] & 0xFFFFFFFFull);
        float obj = d_obj[n];
        float bx = d_bx[n], by = d_by[n], bw = d_bw[n], bh = d_bh[n];
        u.srt.x1[r] = bx - 0.5f * bw;  u.srt.x2[r] = bx + 0.5f * bw;
        u.srt.y1[r] = by - 0.5f * bh;  u.srt.y2[r] = by + 0.5f * bh;
        u.srt.area[r] = bw * bh;
        u.srt.obj[r]  = obj;
        u.srt.s[r]    = obj;
        if (r < NDET && obj > OBJ_T) localcnt++;
    }
    if (localcnt) atomicAdd(&s_cnt, localcnt);   // LDS ds_add
    __syncthreads();
    const int num_obj = s_cnt;

    // ---- Phase D: greedy NMS (sequential over i, parallel over j) ----
    for (int i = 0; i < NDET; ++i) {
        float si = u.srt.s[i];                   // block-uniform LDS read
        bool active = (si > 0.01f) && (u.srt.obj[i] > OBJ_T);
        if (active) {                            // uniform branch
            float xi1 = u.srt.x1[i], xi2 = u.srt.x2[i];
            float yi1 = u.srt.y1[i], yi2 = u.srt.y2[i];
            float ai  = u.srt.area[i];
            for (int q = i + 1 + tid; q < NPAD; q += NTHREADS) {
                float iw = fminf(u.srt.x2[q], xi2) - fmaxf(u.srt.x1[q], xi1);
                float ih = fminf(u.srt.y2[q], yi2) - fmaxf(u.srt.y1[q], yi1);
                iw = fmaxf(iw, 0.0f);
                ih = fmaxf(ih, 0.0f);
                float inter = iw * ih;
                float iou = inter / (u.srt.area[q] + ai - inter);
                if (iou >= NMS_T) u.srt.s[q] = 0.0f;
            }
            __syncthreads();                     // only active iterations wrote
        }
        // inactive iterations perform no writes -> no barrier needed
    }

    // ---- Phase E: final filter + masked write ----
    float* outb = out + (size_t)b * NDET * 7;
    for (int r = tid; r < NDET; r += NTHREADS) {
        int n = (NPAD - 1) - (int)(unsigned)(keys[r] & 0xFFFFFFFFull);
        float obj = d_obj[n];
        bool valid = obj > OBJ_T;
        bool keep  = valid && (u.srt.s[r] > 0.01f) && (r != num_obj - 1);
        float prob = d_prob[n];
        float p    = (prob > CLS_T) ? prob : 0.0f;
        float bw   = d_bw[n];
        bool fin   = keep && (p > 0.01f) && (bw * bw > 0.0004f);
        float m    = fin ? 1.0f : 0.0f;

        float* o = outb + (size_t)r * 7;
        o[0] = obj * m;
        o[1] = d_bx[n] * m;
        o[2] = d_by[n] * m;
        o[3] = bw * m;
        o[4] = d_bh[n] * m;
        o[5] = d_cid[n] * m;
        o[6] = p * m;
    }
}

extern "C" void kernel_launch(void* const* d_in, const int* in_sizes, int n_in,
                              void* d_out, int out_size, void* d_ws, size_t ws_size,
                              hipStream_t stream) {
    (void)n_in; (void)out_size; (void)d_ws; (void)ws_size;
    const float* pred    = (const float*)d_in[0];
    const float* anchors = (const float*)d_in[1];
    float* out = (float*)d_out;
    int batch = in_sizes[0] / (NUMA * ALEN * HW);   // 128
    yolov2_fused_kernel<<<batch, NTHREADS, 0, stream>>>(pred, anchors, out);
}